// RED_GNN_7816840479042
// MI455X (gfx1250) — compile-verified
//
#include <hip/hip_runtime.h>
#include <cstddef>
#include <cstdint>

// ---------------------------------------------------------------------------
// RED-GNN layer, refactored for MI455X (gfx1250):
//   WsH  = hidden @ Ws.T                    (WMMA f32 16x16x4, N x 64)
//   WrR  = rela_embed @ Wr.T                (tiny table, VALU)
//   WqrQ = rela_embed[q_rel] @ Wqr.T + b    (tiny table, VALU)
//   per edge: alpha = sigmoid(wa . relu(WsH[s]+WrR[r]+WqrQ[q]) + wa_b)
//             agg[obj] += alpha * (hidden[s] + rela_embed[r])   (f32 atomics)
//   out  = agg @ Wh.T                       (WMMA f32 16x16x4)
// ---------------------------------------------------------------------------

typedef __attribute__((ext_vector_type(2))) float v2f;
typedef __attribute__((ext_vector_type(8))) float v8f;

// ---------- tiny table kernels (compute-trivial) ----------
__global__ __launch_bounds__(256) void rel_table_kernel(
    const float* __restrict__ rela,  // [R,64]
    const float* __restrict__ Wr,    // [64,64]
    float* __restrict__ out,         // [R,64]
    int R) {
  int t = blockIdx.x * blockDim.x + threadIdx.x;
  if (t >= R * 64) return;
  int r = t >> 6, d = t & 63;
  const float* g = rela + (size_t)r * 64;
  const float* w = Wr + (size_t)d * 64;
  float acc = 0.f;
#pragma unroll 8
  for (int k = 0; k < 64; ++k) acc = fmaf(g[k], w[k], acc);
  out[t] = acc;
}

__global__ __launch_bounds__(256) void qr_table_kernel(
    const float* __restrict__ rela,   // [R,64]
    const int* __restrict__ q_rel,    // [NQ]
    const float* __restrict__ Wq,     // [64,64]
    const float* __restrict__ bq,     // [64]
    float* __restrict__ out,          // [NQ,64]
    int NQ) {
  int t = blockIdx.x * blockDim.x + threadIdx.x;
  if (t >= NQ * 64) return;
  int q = t >> 6, d = t & 63;
  const float* g = rela + (size_t)q_rel[q] * 64;
  const float* w = Wq + (size_t)d * 64;
  float acc = bq[d];
#pragma unroll 8
  for (int k = 0; k < 64; ++k) acc = fmaf(g[k], w[k], acc);
  out[t] = acc;
}

// ---------- Y[rows,64] = X[rows,64] @ W[64,64].T via V_WMMA_F32_16X16X4_F32 --
// One wave (32 threads) per 16-row tile. 4 N-tiles of 16 cols, 16 K-steps of 4.
// A 16x4 f32 layout: lanes 0-15 -> M=lane, VGPR0/1 = K=0/1; lanes 16-31 K=2/3.
// B 4x16 f32 layout mirrors with N across lanes.  C/D: 8 VGPRs, M=r(+8), N=lane.
__global__ __launch_bounds__(32) void gemm_xwt_f32_wmma(
    const float* __restrict__ X, const float* __restrict__ W,
    float* __restrict__ Y, int rows) {
  const int lane  = threadIdx.x & 31;
  const int m0    = blockIdx.x << 4;       // 16-row tile base (scalar)
  const int mlane = lane & 15;
  const int khalf = lane >> 4;             // 0: K=0,1  1: K=2,3
  int mrow = m0 + mlane;
  if (mrow > rows - 1) mrow = rows - 1;    // clamp (no divergence, no branch)
  const float* xrow = X + (size_t)mrow * 64;

  v8f c[4] = {v8f{0.f}, v8f{0.f}, v8f{0.f}, v8f{0.f}};

#pragma unroll
  for (int k = 0; k < 64; k += 4) {
    const int ko = k + khalf * 2;
    v2f a;
    a.x = xrow[ko];
    a.y = xrow[ko + 1];
#pragma unroll
    for (int nt = 0; nt < 4; ++nt) {
      const float* wrow = W + (size_t)(nt * 16 + mlane) * 64;  // B[k][n]=W[n][k]
      v2f b;
      b.x = wrow[ko];
      b.y = wrow[ko + 1];
      c[nt] = __builtin_amdgcn_wmma_f32_16x16x4_f32(
          false, a, false, b, (short)0, c[nt], false, false);
    }
  }

#pragma unroll
  for (int nt = 0; nt < 4; ++nt) {
#pragma unroll
    for (int r = 0; r < 8; ++r) {
      int row = m0 + r + 8 * khalf;
      if (row < rows) Y[(size_t)row * 64 + nt * 16 + mlane] = c[nt][r];
    }
  }
}

// ---------- per-edge attention + scatter-add (memory-bound) ----------------
// 16 lanes per edge, each lane owns 4 consecutive floats (float4).
__global__ __launch_bounds__(256) void edge_kernel(
    const float4* __restrict__ hidden4,  // [N,16]
    const float4* __restrict__ rela4,    // [R,16]
    const float4* __restrict__ WsH4,     // [N,16]
    const float4* __restrict__ WrR4,     // [R,16]
    const float4* __restrict__ WqrQ4,    // [NQ,16]
    const float4* __restrict__ wa4,      // [16]
    const float* __restrict__ wa_b,      // [1]
    const int* __restrict__ sub, const int* __restrict__ rel,
    const int* __restrict__ r_idx, const int* __restrict__ obj,
    float* __restrict__ agg,             // [N,64] pre-zeroed
    int n_edge) {
  int t = blockIdx.x * blockDim.x + threadIdx.x;
  int e = t >> 4;
  if (e >= n_edge) return;
  int l = t & 15;

  int s = sub[e], r = rel[e], q = r_idx[e], o = obj[e];

  float4 ps = WsH4[(size_t)s * 16 + l];
  float4 pr = WrR4[(size_t)r * 16 + l];
  float4 pq = WqrQ4[(size_t)q * 16 + l];
  float x0 = fmaxf(ps.x + pr.x + pq.x, 0.f);
  float x1 = fmaxf(ps.y + pr.y + pq.y, 0.f);
  float x2 = fmaxf(ps.z + pr.z + pq.z, 0.f);
  float x3 = fmaxf(ps.w + pr.w + pq.w, 0.f);

  float4 w = wa4[l];
  float part = x0 * w.x + x1 * w.y + x2 * w.z + x3 * w.w;
  // reduce across the 16 lanes of this edge (xor masks 1..8 stay in-group)
#pragma unroll
  for (int m = 1; m < 16; m <<= 1) part += __shfl_xor(part, m, 32);
  float alpha = 1.0f / (1.0f + __expf(-(part + wa_b[0])));

  float4 h = hidden4[(size_t)s * 16 + l];
  float4 hr = rela4[(size_t)r * 16 + l];
  float* dst = agg + (size_t)o * 64 + l * 4;
  unsafeAtomicAdd(dst + 0, alpha * (h.x + hr.x));
  unsafeAtomicAdd(dst + 1, alpha * (h.y + hr.y));
  unsafeAtomicAdd(dst + 2, alpha * (h.z + hr.z));
  unsafeAtomicAdd(dst + 3, alpha * (h.w + hr.w));
}

extern "C" void kernel_launch(void* const* d_in, const int* in_sizes, int n_in,
                              void* d_out, int out_size, void* d_ws, size_t ws_size,
                              hipStream_t stream) {
  const float* hidden = (const float*)d_in[0];
  const int*   q_rel  = (const int*)d_in[1];
  const int*   r_idx  = (const int*)d_in[2];
  const int*   rel    = (const int*)d_in[3];
  const int*   sub    = (const int*)d_in[4];
  const int*   obj    = (const int*)d_in[5];
  // d_in[6] = n_node scalar (device); derive N from sizes instead
  const float* rela   = (const float*)d_in[7];
  const float* Ws     = (const float*)d_in[8];
  const float* Wr     = (const float*)d_in[9];
  const float* WqrW   = (const float*)d_in[10];
  const float* Wqrb   = (const float*)d_in[11];
  const float* waW    = (const float*)d_in[12];
  const float* wab    = (const float*)d_in[13];
  const float* Wh     = (const float*)d_in[14];

  const int N  = in_sizes[0] / 64;   // 500000
  const int NQ = in_sizes[1];        // 5000
  const int E  = in_sizes[4];        // 1000000
  const int R  = in_sizes[7] / 64;   // 401

  // workspace carve-out (256B aligned slabs)
  char* ws = (char*)d_ws;
  size_t off = 0;
  auto carve = [&](size_t bytes) {
    void* p = ws + off;
    off += (bytes + 255) & ~(size_t)255;
    return p;
  };
  float* WsH  = (float*)carve((size_t)N * 64 * sizeof(float));
  float* agg  = (float*)carve((size_t)N * 64 * sizeof(float));
  float* WrR  = (float*)carve((size_t)R * 64 * sizeof(float));
  float* WqrQ = (float*)carve((size_t)NQ * 64 * sizeof(float));
  (void)ws_size; (void)n_in; (void)out_size;

  hipMemsetAsync(agg, 0, (size_t)N * 64 * sizeof(float), stream);

  rel_table_kernel<<<(R * 64 + 255) / 256, 256, 0, stream>>>(rela, Wr, WrR, R);
  qr_table_kernel<<<(NQ * 64 + 255) / 256, 256, 0, stream>>>(rela, q_rel, WqrW,
                                                             Wqrb, WqrQ, NQ);
  gemm_xwt_f32_wmma<<<(N + 15) / 16, 32, 0, stream>>>(hidden, Ws, WsH, N);

  int edge_threads_blocks = (int)(((size_t)E * 16 + 255) / 256);
  edge_kernel<<<edge_threads_blocks, 256, 0, stream>>>(
      (const float4*)hidden, (const float4*)rela, (const float4*)WsH,
      (const float4*)WrR, (const float4*)WqrQ, (const float4*)waW, wab,
      sub, rel, r_idx, obj, agg, E);

  gemm_xwt_f32_wmma<<<(N + 15) / 16, 32, 0, stream>>>(agg, Wh, (float*)d_out, N);
}